// GraphSAGENet_80676665688561
// MI455X (gfx1250) — compile-verified
//
#include <hip/hip_runtime.h>

typedef __attribute__((ext_vector_type(2))) float v2f;
typedef __attribute__((ext_vector_type(8))) float v8f;

#define N_NODES 50000
#define KDIM 128
#define BN_EPS 1e-5f

// ---------------- utility kernels ----------------

__global__ void zero_f32(float* __restrict__ p, long n) {
  long i = (long)blockIdx.x * blockDim.x + threadIdx.x;
  long stride = (long)gridDim.x * blockDim.x;
  for (; i < n; i += stride) p[i] = 0.0f;
}

__global__ void degree_kernel(const int* __restrict__ dst, float* __restrict__ deg, int E) {
  int t = blockIdx.x * blockDim.x + threadIdx.x;
  if (t < E) atomicAdd(&deg[dst[t]], 1.0f);
}

__global__ void invdeg_kernel(float* __restrict__ deg, int n) {
  int i = blockIdx.x * blockDim.x + threadIdx.x;
  if (i < n) deg[i] = 1.0f / fmaxf(deg[i], 1.0f);
}

// one wave (32 lanes) per edge: 4 contiguous floats per lane (512B coalesced gather),
// 4 f32 atomic adds per lane (global_atomic_add_f32, no return -> STOREcnt path)
__global__ void scatter_kernel(const float* __restrict__ x, const int* __restrict__ src,
                               const int* __restrict__ dst, float* __restrict__ agg, int E) {
  int t = blockIdx.x * blockDim.x + threadIdx.x;
  int e = t >> 5;
  int lane = t & 31;
  if (e >= E) return;
  int s = src[e];
  int d = dst[e];
  const float4 v = *(const float4*)(x + (size_t)s * KDIM + lane * 4);
  float* o = agg + (size_t)d * KDIM + lane * 4;
  atomicAdd(o + 0, v.x);
  atomicAdd(o + 1, v.y);
  atomicAdd(o + 2, v.z);
  atomicAdd(o + 3, v.w);
}

// ---------------- WMMA GEMM ----------------
// C[N, DOUT] (op)= (rowscale? * A[N,128]) @ W[DOUT,128]^T
// MODE 0: C = diag(rowscale) * acc      (SAGE agg pass; scale folded into epilogue)
// MODE 1: C = relu(bn(C + acc + bias))  (layer-1 second pass, fused BatchNorm+ReLU)
// MODE 2: C = C + acc + bias            (layer-2 second pass)
//
// No LDS: B fragments {B[ke][col], B[ke+1][col]} are the contiguous pair
// W[col][ke..ke+1] in the row-major weight matrix -> one global_load_b64 per
// K-step from a loop-invariant base with an immediate offset. W (<=64 KB) is
// read by every wave and stays WGP$/L2 resident, so total global traffic is
// identical to an LDS-staged scheme but with zero per-iteration address VALU.
// N % 16 == 0, so EXEC is all-ones around every WMMA.

template <int DOUT, int MODE>
__global__ __launch_bounds__(DOUT * 2)
void gemm16_wmma(const float* __restrict__ A, const float* __restrict__ rowscale,
                 const float* __restrict__ W, const float* __restrict__ bias,
                 const float* __restrict__ gam, const float* __restrict__ bet,
                 const float* __restrict__ mu, const float* __restrict__ var,
                 float* __restrict__ C) {
  const int wave = threadIdx.x >> 5;
  const int lane = threadIdx.x & 31;
  const int grp  = lane >> 4;   // 0: K pair {0,1}, 1: K pair {2,3}
  const int mr   = lane & 15;
  const int row0 = blockIdx.x * 16;
  const int col  = wave * 16 + mr;

  // A fragment: lane feeds row (row0+mr), K offset grp*2
  const float* ap = A + (size_t)(row0 + mr) * KDIM + grp * 2;
  // B fragment: lane feeds column `col`, K offset grp*2 (contiguous in W row)
  const float* wp = W + (size_t)col * KDIM + grp * 2;

  v8f acc = {};
#pragma unroll
  for (int k0 = 0; k0 < KDIM; k0 += 4) {
    v2f a = *(const v2f*)(ap + k0);   // global_load_b64, immediate offset
    v2f b = *(const v2f*)(wp + k0);   // global_load_b64, immediate offset
    // D = A(16x4) * B(4x16) + C, full f32
    acc = __builtin_amdgcn_wmma_f32_16x16x4_f32(false, a, false, b, (short)0, acc,
                                                false, false);
  }

  // epilogue: VGPR r holds M = row0 + r + 8*grp, N = col
  const int rbase = row0 + grp * 8;
  float bv = 0.0f, bn_scale = 0.0f, bn_shift = 0.0f;
  if (MODE >= 1) bv = bias[col];
  if (MODE == 1) {
    bn_scale = gam[col] * rsqrtf(var[col] + BN_EPS);
    bn_shift = bet[col] - mu[col] * bn_scale;
  }
  float* crow = C + (size_t)rbase * DOUT + col;
#pragma unroll
  for (int r = 0; r < 8; ++r) {
    float v = acc[r];
    float* cp = crow + (size_t)r * DOUT;
    if (MODE == 0) {
      v *= rowscale[rbase + r];   // mean-normalization folded into epilogue
      *cp = v;
    } else {
      v += *cp + bv;
      if (MODE == 1) {
        v = v * bn_scale + bn_shift;
        v = fmaxf(v, 0.0f);
      }
      *cp = v;
    }
  }
}

// ---------------- launcher ----------------

extern "C" void kernel_launch(void* const* d_in, const int* in_sizes, int n_in,
                              void* d_out, int out_size, void* d_ws, size_t ws_size,
                              hipStream_t stream) {
  const float* x   = (const float*)d_in[0];
  const int*   ei  = (const int*)d_in[1];   // edge_index [2, E] (int32 under jax defaults)
  const float* W1l = (const float*)d_in[2];
  const float* b1l = (const float*)d_in[3];
  const float* W1r = (const float*)d_in[4];
  const float* gam = (const float*)d_in[5];
  const float* bet = (const float*)d_in[6];
  const float* mu  = (const float*)d_in[7];
  const float* var = (const float*)d_in[8];
  const float* W2l = (const float*)d_in[9];
  const float* b2l = (const float*)d_in[10];
  const float* W2r = (const float*)d_in[11];
  float* out = (float*)d_out;

  const int N = N_NODES;
  const int E = in_sizes[1] / 2;
  const int* src = ei;
  const int* dst = ei + E;

  // workspace layout
  char* ws = (char*)d_ws;
  float* deg = (float*)ws;                                         // N floats (-> inv_deg)
  size_t off = ((size_t)N * sizeof(float) + 255) & ~(size_t)255;
  float* agg = (float*)(ws + off);                                 // N*128 floats
  float* h   = agg + (size_t)N * KDIM;                             // N*128 floats

  const long aggN = (long)N * KDIM;
  const int scatter_blocks = (int)(((long)E * 32 + 255) / 256);

  // ---- layer 1 ----
  zero_f32<<<2048, 256, 0, stream>>>(deg, N);
  zero_f32<<<8192, 256, 0, stream>>>(agg, aggN);
  degree_kernel<<<(E + 255) / 256, 256, 0, stream>>>(dst, deg, E);
  invdeg_kernel<<<(N + 255) / 256, 256, 0, stream>>>(deg, N);
  scatter_kernel<<<scatter_blocks, 256, 0, stream>>>(x, src, dst, agg, E);

  gemm16_wmma<128, 0><<<N / 16, 256, 0, stream>>>(agg, deg, W1l, nullptr, nullptr,
                                                  nullptr, nullptr, nullptr, h);
  gemm16_wmma<128, 1><<<N / 16, 256, 0, stream>>>(x, nullptr, W1r, b1l, gam, bet,
                                                  mu, var, h);

  // ---- layer 2 ----
  zero_f32<<<8192, 256, 0, stream>>>(agg, aggN);
  scatter_kernel<<<scatter_blocks, 256, 0, stream>>>(h, src, dst, agg, E);

  gemm16_wmma<64, 0><<<N / 16, 128, 0, stream>>>(agg, deg, W2l, nullptr, nullptr,
                                                 nullptr, nullptr, nullptr, out);
  gemm16_wmma<64, 2><<<N / 16, 128, 0, stream>>>(h, nullptr, W2r, b2l, nullptr,
                                                 nullptr, nullptr, nullptr, out);
}